// HybridAttention_69011534512638
// MI455X (gfx1250) — compile-verified
//
#include <hip/hip_runtime.h>
#include <math.h>

// ---------------------------------------------------------------------------
// HybridAttention for MI455X (gfx1250, wave32).
//
// Roofline: context tensors = 536 MB f32 @ 23.3 TB/s => ~23 us floor if each
// context element is read exactly once. The attention core is therefore a
// fused online-softmax (flash) pass with context held in VGPRs for both the
// score dot-product and the weighted accumulation. The dense GEMMs
// (inputs@W_in.T, combined@W_out.T) use V_WMMA_F32_16X16X4_F32 (f32 in/out,
// no precision downgrade; GEMM FLOPs are negligible vs the memory floor).
// Both branches (tree/txt) are independent -> run concurrently via gridDim.y
// to maximize memory-level parallelism on the bandwidth-bound phase.
// ---------------------------------------------------------------------------

typedef float v2f __attribute__((ext_vector_type(2)));
typedef float v8f __attribute__((ext_vector_type(8)));

// ---------------------------------------------------------------------------
// C[M,N] = A[M,K] @ Bm[N,K]^T, optional tanh epilogue.
// blockIdx.y selects the (A,C) pair so independent GEMMs run concurrently.
// One 16x16 output tile per wave; 8 waves per 256-thread block.
// WMMA f32 16x16x4 fragment layout (ISA 7.12.2):
//   A (16x4):  lane&15 = M row, lane>=16 => K-offset +2; v2f = {K, K+1}
//   B (4x16):  lane&15 = N col, lane>=16 => K-offset +2; v2f = {K, K+1}
//   C (16x16): VGPR r: lanes 0-15 -> M=r, lanes 16-31 -> M=r+8; N = lane&15
// ---------------------------------------------------------------------------
__global__ __launch_bounds__(256) void gemm_nt_wmma_f32(
    const float* __restrict__ A0,  // [M,K] branch 0
    const float* __restrict__ A1,  // [M,K] branch 1
    const float* __restrict__ Bm,  // [N,K] shared weights
    float*       __restrict__ C0,  // [M,N] branch 0
    float*       __restrict__ C1,  // [M,N] branch 1
    int M, int N, int K, int apply_tanh)
{
  const float* A = blockIdx.y ? A1 : A0;
  float*       C = blockIdx.y ? C1 : C0;

  const int lane   = threadIdx.x & 31;
  const int wave   = threadIdx.x >> 5;
  const int ntiles = N >> 4;
  const int tile   = blockIdx.x * 8 + wave;
  if (tile >= (M >> 4) * ntiles) return;   // wave-uniform: EXEC stays all-1s

  const int mt = tile / ntiles;
  const int mb = mt << 4;
  const int nb = (tile - mt * ntiles) << 4;
  const int half = lane >> 4;   // 0: K,K+1   1: K+2,K+3
  const int l15  = lane & 15;

  const float* arow = A  + (size_t)(mb + l15) * K + half * 2;
  const float* brow = Bm + (size_t)(nb + l15) * K + half * 2;

  v8f c = {};
#pragma unroll 4
  for (int k = 0; k < K; k += 4) {
    v2f a = *(const v2f*)(arow + k);
    v2f b = *(const v2f*)(brow + k);
    // (neg_a, A, neg_b, B, c_mod, C, reuse_a, reuse_b)
    c = __builtin_amdgcn_wmma_f32_16x16x4_f32(
        false, a, false, b, (short)0, c, false, false);
  }

#pragma unroll
  for (int r = 0; r < 8; ++r) {
    const int row = mb + r + half * 8;
    const int col = nb + l15;
    float v = c[r];
    if (apply_tanh) v = tanhf(v);
    C[(size_t)row * N + col] = v;
  }
}

// ---------------------------------------------------------------------------
// Fused flash attention. Grid = (B, 2): blockIdx.x = batch, blockIdx.y =
// branch (0=tree, 1=txt). 256 threads = 8 waves; each wave owns s = wave,
// wave+8, ...; each lane owns 32 of the D=1024 channels (8x float4), holding
// target / context-row / accumulator in VGPRs so context streams from HBM
// exactly once. Raw masked scores go to the attn output region (finalized
// later). Weighted context + a copy of `inputs` land in `combined` [B,4D].
// ---------------------------------------------------------------------------
__global__ __launch_bounds__(256) void flash_attn_fused(
    const float*         __restrict__ inputs0,
    const float*         __restrict__ inputs1,
    const float*         __restrict__ context0,
    const float*         __restrict__ context1,
    const unsigned char* __restrict__ mask0,
    const unsigned char* __restrict__ mask1,
    const float*         __restrict__ target0,
    const float*         __restrict__ target1,
    float*               __restrict__ scores0,   // [B,S] raw masked scores
    float*               __restrict__ scores1,
    float*               __restrict__ stats0,    // [B,2] (row max, row sum)
    float*               __restrict__ stats1,
    float*               __restrict__ combined,  // [B,4D]
    int S, int D)
{
  constexpr int NW = 8;
  const int branch = blockIdx.y;
  const float*         inputs  = branch ? inputs1  : inputs0;
  const float*         context = branch ? context1 : context0;
  const unsigned char* mask    = branch ? mask1    : mask0;
  const float*         target  = branch ? target1  : target0;
  float*               scores  = branch ? scores1  : scores0;
  float*               stats   = branch ? stats1   : stats0;
  const int wc_off = branch * 2 * D;   // weighted-context slot in combined
  const int in_off = wc_off + D;       // inputs-copy slot in combined

  const int b    = blockIdx.x;
  const int tid  = threadIdx.x;
  const int wave = tid >> 5;
  const int lane = tid & 31;
  const int J    = D >> 7;             // float4 chunks per lane (8 @ D=1024)

  __shared__ float sm_m[NW];
  __shared__ float sm_s[NW];
  __shared__ float sm_acc[1024];

  for (int d = tid; d < D; d += 256) sm_acc[d] = 0.0f;

  const float* ctxb = context + (size_t)b * S * D;
  const float* tgtb = target  + (size_t)b * D;

  float4 tgt[8], acc[8];
#pragma unroll
  for (int j = 0; j < 8; ++j) {
    if (j < J) tgt[j] = *(const float4*)(tgtb + lane * 4 + 128 * j);
    acc[j] = make_float4(0.f, 0.f, 0.f, 0.f);
  }

  float m = -__builtin_inff();
  float sum = 0.0f;

  for (int s = wave; s < S; s += NW) {
    const float* crow = ctxb + (size_t)s * D;
    // Hint next row for this wave into cache (global_prefetch_b8).
    __builtin_prefetch(crow + (size_t)NW * D + lane * 4, 0, 0);

    float4 cx[8];
    float part = 0.0f;
#pragma unroll
    for (int j = 0; j < 8; ++j) {
      if (j < J) {
        cx[j] = *(const float4*)(crow + lane * 4 + 128 * j);
        part += cx[j].x * tgt[j].x + cx[j].y * tgt[j].y
              + cx[j].z * tgt[j].z + cx[j].w * tgt[j].w;
      }
    }
    // wave32 xor-reduction: every lane ends with the full dot product
#pragma unroll
    for (int off = 16; off > 0; off >>= 1)
      part += __shfl_xor(part, off, 32);

    const float score = mask[(size_t)b * S + s] ? -__builtin_inff() : part;
    if (lane == 0) scores[(size_t)b * S + s] = score;

    const float nm = fmaxf(m, score);
    if (nm != -__builtin_inff()) {          // skip while everything is masked
      const float corr = __expf(m - nm);    // m=-inf -> 0
      const float p    = __expf(score - nm);
      sum = sum * corr + p;
#pragma unroll
      for (int j = 0; j < 8; ++j) {
        if (j < J) {
          acc[j].x = acc[j].x * corr + p * cx[j].x;
          acc[j].y = acc[j].y * corr + p * cx[j].y;
          acc[j].z = acc[j].z * corr + p * cx[j].z;
          acc[j].w = acc[j].w * corr + p * cx[j].w;
        }
      }
      m = nm;
    }
  }

  if (lane == 0) { sm_m[wave] = m; sm_s[wave] = sum; }
  __syncthreads();

  float M = sm_m[0];
#pragma unroll
  for (int w = 1; w < NW; ++w) M = fmaxf(M, sm_m[w]);
  float total = 0.0f;
#pragma unroll
  for (int w = 0; w < NW; ++w)
    if (sm_m[w] != -__builtin_inff()) total += sm_s[w] * __expf(sm_m[w] - M);

  const float factor = (m == -__builtin_inff()) ? 0.0f : __expf(m - M);

  // cross-wave combine via LDS float atomics (ds_add_f32)
#pragma unroll
  for (int j = 0; j < 8; ++j) {
    if (j < J) {
      const int d = lane * 4 + 128 * j;
      atomicAdd(&sm_acc[d + 0], factor * acc[j].x);
      atomicAdd(&sm_acc[d + 1], factor * acc[j].y);
      atomicAdd(&sm_acc[d + 2], factor * acc[j].z);
      atomicAdd(&sm_acc[d + 3], factor * acc[j].w);
    }
  }
  __syncthreads();

  const float inv = 1.0f / total;
  float* comb = combined + (size_t)b * (4 * D);
  for (int d = tid; d < D; d += 256) {
    comb[wc_off + d] = sm_acc[d] * inv;
    comb[in_off + d] = inputs[(size_t)b * D + d];
  }
  if (tid == 0) { stats[2 * b] = M; stats[2 * b + 1] = total; }
}

// attn = exp(score - M) / total  (masked scores are -inf -> exp -> 0).
// Covers BOTH branches in one launch: scores/stats are contiguous
// branch-major, so rows = 2*B.
__global__ __launch_bounds__(256) void softmax_finalize(
    float* __restrict__ scores, const float* __restrict__ stats,
    int rows, int S)
{
  const int i = blockIdx.x * blockDim.x + threadIdx.x;
  if (i >= rows * S) return;
  const int b = i / S;
  const float M = stats[2 * b];
  const float T = stats[2 * b + 1];
  scores[i] = __expf(scores[i] - M) / T;
}

// ---------------------------------------------------------------------------
extern "C" void kernel_launch(void* const* d_in, const int* in_sizes, int n_in,
                              void* d_out, int out_size, void* d_ws, size_t ws_size,
                              hipStream_t stream) {
  const float*         inputs_tree  = (const float*)d_in[0];
  const float*         context_tree = (const float*)d_in[1];
  const float*         inputs_txt   = (const float*)d_in[2];
  const float*         context_txt  = (const float*)d_in[3];
  const unsigned char* mask_tree    = (const unsigned char*)d_in[4];
  const unsigned char* mask_txt     = (const unsigned char*)d_in[5];
  const float*         W_in         = (const float*)d_in[6];
  const float*         W_out        = (const float*)d_in[7];

  // Shapes: in_sizes[0]=B*D, in_sizes[1]=B*S*D, in_sizes[6]=D*D
  const int S = in_sizes[1] / in_sizes[0];
  int D = 1;
  while ((long long)(D + 1) * (D + 1) <= (long long)in_sizes[6]) ++D;
  const int B = in_sizes[0] / D;

  // Workspace layout (floats); stats arrays contiguous branch-major.
  float* ws         = (float*)d_ws;
  float* tgt_tree   = ws;                           // B*D
  float* tgt_txt    = tgt_tree + (size_t)B * D;     // B*D
  float* combined   = tgt_txt  + (size_t)B * D;     // B*4D
  float* stats_tree = combined + (size_t)B * 4 * D; // B*2
  float* stats_txt  = stats_tree + 2 * B;           // B*2

  // Output layout: out [B,D] | attn_tree [B,S] | attn_txt [B,S] (contiguous)
  float* out       = (float*)d_out;
  float* attn_tree = out + (size_t)B * D;
  float* attn_txt  = attn_tree + (size_t)B * S;

  const int tiles = (B >> 4) * (D >> 4);
  const int gblks = (tiles + 7) / 8;

  // 1) target = inputs @ W_in.T for BOTH branches concurrently (gridDim.y=2)
  gemm_nt_wmma_f32<<<dim3(gblks, 2), 256, 0, stream>>>(
      inputs_tree, inputs_txt, W_in, tgt_tree, tgt_txt, B, D, D, 0);

  // 2) fused flash attention, both branches concurrently: grid (B, 2)
  flash_attn_fused<<<dim3(B, 2), 256, 0, stream>>>(
      inputs_tree, inputs_txt, context_tree, context_txt,
      mask_tree, mask_txt, tgt_tree, tgt_txt,
      attn_tree, attn_txt, stats_tree, stats_txt,
      combined, S, D);

  // 3) finalize attn = softmax(scores), both branches in one launch
  const int fin_blks = (2 * B * S + 255) / 256;
  softmax_finalize<<<fin_blks, 256, 0, stream>>>(
      attn_tree, stats_tree, 2 * B, S);

  // 4) out = tanh(combined @ W_out.T)   (WMMA f32 16x16x4, K = 4D)
  gemm_nt_wmma_f32<<<dim3(gblks, 1), 256, 0, stream>>>(
      combined, combined, W_out, out, out, B, D, 4 * D, 1);
}